// RelativePositionEncodingWithIndexRemoval_60017872994823
// MI455X (gfx1250) — compile-verified
//
#include <hip/hip_runtime.h>
#include <hip/hip_bf16.h>
#include <stdint.h>

// Problem constants (match reference)
#define R_MAX 32
#define S_MAX 2
#define NTB   (2 * R_MAX + 2 + 1)               // 67
#define CZ    128
#define IN_DIM (2 * NTB + (2 * S_MAX + 2) + 1)  // 141
#define WELEMS (IN_DIM * CZ)                    // 18048 floats = 72192 bytes
#define BLOCK 256

typedef float    v4f __attribute__((ext_vector_type(4)));
typedef unsigned v4u __attribute__((ext_vector_type(4)));
typedef int      v4i __attribute__((ext_vector_type(4)));
typedef int      v8i __attribute__((ext_vector_type(8)));

__global__ __launch_bounds__(BLOCK)
void relpos_embed_kernel(const int* __restrict__ asym_id,
                         const int* __restrict__ entity_id,
                         const int* __restrict__ residue_index,
                         const int* __restrict__ token_index,
                         const int* __restrict__ sym_id,
                         const int* __restrict__ pair_mask,
                         const float* __restrict__ W,
                         float* __restrict__ out,
                         int N)
{
    // 141 x 128 fp32 table staged in LDS (72 KB; WGP has 320 KB)
    __shared__ float smem[WELEMS];

    const int tid = threadIdx.x;
    const int i   = blockIdx.x;     // query row: one block per i

    // ---------------- Stage W into LDS via the Tensor Data Mover ----------------
#if __has_builtin(__builtin_amdgcn_tensor_load_to_lds)
    if (tid < 32) {  // one wave issues the TDM descriptor (scalar operands)
        const unsigned lds_off = (unsigned)(uintptr_t)&smem[0]; // flat->LDS offset = low 32 bits
        const uint64_t ga      = (uint64_t)(uintptr_t)W;
        const unsigned D       = (unsigned)WELEMS;              // 18048 elems of 4B

        v4u g0;
        g0.x = 1u;                                              // count=1, user descriptor
        g0.y = lds_off;                                         // lds_addr [63:32]
        g0.z = (unsigned)(ga & 0xffffffffu);                    // global_addr [95:64]
        g0.w = (unsigned)((ga >> 32) & 0x01ffffffu) | (2u << 30); // addr[56:32] | type=2

        v8i g1;
        g1[0] = (int)(2u << 16);                                // data_size = 4 bytes
        g1[1] = (int)((D & 0xffffu) << 16);                     // tensor_dim0 lo16
        g1[2] = (int)(((D >> 16) & 0xffffu) | (1u << 16));      // dim0 hi16 | tensor_dim1=1 lo16
        g1[3] = (int)((D & 0xffffu) << 16);                     // dim1 hi16=0 | tile_dim0 = D
        g1[4] = 1;                                              // tile_dim1 = 1, tile_dim2 = 0
        g1[5] = (int)D;                                         // tensor_dim0_stride lo32
        g1[6] = (int)((D & 0xffffu) << 16);                     // stride hi16=0 | dim1_stride lo16
        g1[7] = 0;                                              // dim1_stride hi32

        v4i g2 = {0, 0, 0, 0};
        v4i g3 = {0, 0, 0, 0};
#if defined(__clang_major__) && __clang_major__ >= 23
        v8i g1b = {0, 0, 0, 0, 0, 0, 0, 0};
        __builtin_amdgcn_tensor_load_to_lds(g0, g1, g2, g3, g1b, 0);
#else
        __builtin_amdgcn_tensor_load_to_lds(g0, g1, g2, g3, 0);
#endif
#if __has_builtin(__builtin_amdgcn_s_wait_tensorcnt)
        __builtin_amdgcn_s_wait_tensorcnt(0);
#else
        asm volatile("s_wait_tensorcnt 0" ::: "memory");
#endif
    }
#endif
    // Correctness-guaranteed cooperative copy (writes identical bytes; benign
    // overlap with the TDM DMA; removable once TDM is hardware-validated).
    for (int k = tid; k < WELEMS; k += BLOCK) smem[k] = W[k];
    __syncthreads();

    // ---------------- Per-block scalars for row i ----------------
    const int lane = tid & 31;
    const int wave = tid >> 5;
    const int c4   = lane << 2;                      // this lane's 4 channels

    const v4f went = *(const v4f*)&smem[(2 * NTB) * CZ + c4];  // w_ent (loop invariant)

    const int a_i = asym_id[i];
    const int e_i = entity_id[i];
    const int r_i = residue_index[i];
    const int t_i = token_index[i];
    const int s_i = sym_id[i];
    const int* __restrict__ mrow = pair_mask + (size_t)i * N;
    float*     __restrict__ orow = out + (size_t)i * (size_t)N * CZ;

    // Each wave owns 32-wide strips of j; lanes precompute bins for 32 j's,
    // then broadcast one packed bin word per pair via ds_bpermute (__shfl).
    for (int jb = wave << 5; jb < N; jb += BLOCK) {
        const int j = jb + lane;
        int pack = 0;
        if (j < N) {
            const int aj = asym_id[j], ej = entity_id[j];
            const int rj = residue_index[j], tj = token_index[j], sj = sym_id[j];
            const bool same_chain  = (a_i == aj);
            const bool same_entity = (e_i == ej);

            int dr = r_i - rj + R_MAX;
            dr = dr < 0 ? 0 : (dr > 2 * R_MAX ? 2 * R_MAX : dr);
            int d_res = same_chain ? dr : (2 * R_MAX + 1);

            int dt = t_i - tj + R_MAX;
            dt = dt < 0 ? 0 : (dt > 2 * R_MAX ? 2 * R_MAX : dt);
            int d_tok = (same_chain && (r_i == rj)) ? dt : (2 * R_MAX + 1);

            int dc = s_i - sj + S_MAX;
            dc = dc < 0 ? 0 : (dc > 2 * S_MAX ? 2 * S_MAX : dc);
            int d_chain = same_entity ? dc : (2 * S_MAX + 1);

            if (mrow[j] != 0) { d_res = NTB - 1; d_tok = NTB - 1; }

            pack = d_res | (d_tok << 7) | (d_chain << 14) | ((same_entity ? 1 : 0) << 17);
        }

        const int tmax = (N - jb) < 32 ? (N - jb) : 32;
        for (int t = 0; t < tmax; ++t) {
            const int p     = __shfl(pack, t, 32);
            const int d_res = p & 127;
            const int d_tok = (p >> 7) & 127;
            const int d_ch  = (p >> 14) & 7;
            const float ef  = (float)((p >> 17) & 1);

            // All 32 lanes hit the same row at consecutive 16B offsets:
            // 128 consecutive DWORDs -> conflict-free ds_load_b128.
            const v4f vr = *(const v4f*)&smem[d_res * CZ + c4];
            const v4f vt = *(const v4f*)&smem[(NTB + d_tok) * CZ + c4];
            const v4f vc = *(const v4f*)&smem[(2 * NTB + 1 + d_ch) * CZ + c4];

            // Match reference association: ((W_res + W_tok) + ent*w_ent) + W_chain
            const v4f o = ((vr + vt) + ef * went) + vc;

            // Streaming 302 MB output: non-temporal 128-bit stores.
            v4f* dst = (v4f*)(orow + (size_t)(jb + t) * CZ + c4);
            __builtin_nontemporal_store(o, dst);
        }
    }
}

extern "C" void kernel_launch(void* const* d_in, const int* in_sizes, int n_in,
                              void* d_out, int out_size, void* d_ws, size_t ws_size,
                              hipStream_t stream) {
    const int*   asym   = (const int*)d_in[0];
    const int*   entity = (const int*)d_in[1];
    const int*   resi   = (const int*)d_in[2];
    const int*   toki   = (const int*)d_in[3];
    const int*   symi   = (const int*)d_in[4];
    const int*   maskp  = (const int*)d_in[5];
    const float* W      = (const float*)d_in[6];
    float*       out    = (float*)d_out;

    const int N = in_sizes[0];  // B == 1 in the reference setup

    relpos_embed_kernel<<<dim3(N), dim3(BLOCK), 0, stream>>>(
        asym, entity, resi, toki, symi, maskp, W, out, N);
}